// Decoder_46042049413334
// MI455X (gfx1250) — compile-verified
//
#include <hip/hip_runtime.h>
#include <hip/hip_bf16.h>

typedef __attribute__((ext_vector_type(2))) float v2f;
typedef __attribute__((ext_vector_type(8))) float v8f;

__device__ __forceinline__ float sigmf(float x) { return 1.0f / (1.0f + __expf(-x)); }

__device__ __forceinline__ v8f wmma_f32(v2f a, v2f b, v8f c) {
  return __builtin_amdgcn_wmma_f32_16x16x4_f32(false, a, false, b, (short)0, c, false, false);
}

// ---- async copy of a 16x512 f32 tile (row stride 512) into LDS (row stride 516,
// ---- padded so 16-lane b64 fragment reads are LDS-bank-conflict-free).
// ---- Low 32 bits of a flat LDS pointer == LDS byte address (ISA 10.2).
__device__ __forceinline__ void stage16x512(const float* __restrict__ g, float* s)
{
  const int tid = threadIdx.x;           // 128 threads
#pragma unroll
  for (int it = 0; it < 16; ++it) {
    const int gi  = tid + it * 128;      // float4 index within 16x512 tile
    const int row = gi >> 7;
    const int c4  = (gi & 127) << 2;     // float column
    const unsigned lo = (unsigned)(size_t)(s + row * 516 + c4);
    const float* ga = g + (long)row * 512 + c4;
    asm volatile("global_load_async_to_lds_b128 %0, %1, off"
                 :: "v"(lo), "v"(ga) : "memory");
  }
}
__device__ __forceinline__ void wait_async() {
  asm volatile("s_wait_asynccnt 0x0" ::: "memory");
}

// -------- init: zero the grid-barrier state (ws is poisoned with 0xAA) -------
__global__ void zero_u32(unsigned* p, int n) {
  if ((int)threadIdx.x < n) p[threadIdx.x] = 0u;
}

// -------- embedding gather: X[t*B+b, :] = embed[tok[b,t], :] ----------------
__global__ __launch_bounds__(128) void gather_embed(
    const int* __restrict__ tok, const float* __restrict__ embed,
    float* __restrict__ X, int B_, int T_, int D_)
{
  const int row = blockIdx.x;          // row = t*B + b
  const int t = row / B_;
  const int b = row - t * B_;
  const int id = tok[b * T_ + t];
  const float4* src = (const float4*)(embed + (long)id * D_);
  float4* dst = (float4*)(X + (long)row * D_);
  for (int i = threadIdx.x; i < (D_ >> 2); i += blockDim.x) dst[i] = src[i];
}

// -------- generic fp32 WMMA GEMM: C = act(A @ op(B) + bias1 + bias2) --------
// Each wave computes a 64x16 tile (4 m-tiles reuse one B fragment; 4 acc chains).
// Requires M % 64 == 0, N % 16 == 0, K % 4 == 0.
__global__ __launch_bounds__(256) void gemm_wmma(
    const float* __restrict__ A, long lda, long batchA,
    const float* __restrict__ Bm, long ldb, long batchB, int Btrans,
    const float* __restrict__ bias1, const float* __restrict__ bias2,
    float* __restrict__ C, long ldc, long batchC,
    int M, int N, int K, int act)
{
  const int wv   = threadIdx.x >> 5;
  const int lane = threadIdx.x & 31;
  const int half = lane >> 4;
  const int lm   = lane & 15;
  const int ntile = N >> 4;
  const int tile  = blockIdx.x * (blockDim.x >> 5) + wv;
  const int mg = tile / ntile;              // group of 4 m-tiles (64 rows)
  const int nt = tile - mg * ntile;
  if (mg >= (M >> 6)) return;               // wave-uniform
  const long z = blockIdx.z;
  A  += z * batchA;
  Bm += z * batchB;
  C  += z * batchC;
  const int m0 = mg << 6, n0 = nt << 4;

  float bv = 0.0f;
  if (bias1) bv += bias1[n0 + lm];
  if (bias2) bv += bias2[n0 + lm];
  v8f acc[4];
#pragma unroll
  for (int j = 0; j < 4; ++j)
#pragma unroll
    for (int r = 0; r < 8; ++r) acc[j][r] = bv * (j == 0 ? 1.0f : 0.0f) + (j == 0 ? 0.0f : 0.0f), acc[j][r] = (j == 0 ? bv : 0.0f);

  const float* ap = A + (long)(m0 + lm) * lda + 2 * half;
  if (!Btrans) {
    const float* bp = Bm + (long)(n0 + lm) * ldb + 2 * half;
    for (int k = 0; k < K; k += 4) {
      v2f b = *(const v2f*)(bp + k);
#pragma unroll
      for (int j = 0; j < 4; ++j) {
        v2f a = *(const v2f*)(ap + (long)(j << 4) * lda + k);
        acc[j] = wmma_f32(a, b, acc[j]);
      }
    }
  } else {
    const float* bp = Bm + (long)(2 * half) * ldb + (n0 + lm);
    for (int k = 0; k < K; k += 4) {
      v2f b;
      b.x = bp[(long)k * ldb];
      b.y = bp[(long)k * ldb + ldb];
#pragma unroll
      for (int j = 0; j < 4; ++j) {
        v2f a = *(const v2f*)(ap + (long)(j << 4) * lda + k);
        acc[j] = wmma_f32(a, b, acc[j]);
      }
    }
  }

#pragma unroll
  for (int j = 0; j < 4; ++j) {
    float* cp = C + (long)(m0 + (j << 4) + 8 * half) * ldc + n0 + lm;
#pragma unroll
    for (int r = 0; r < 8; ++r) {
      float v = acc[j][r];
      if (act == 1) v = tanhf(v);
      cp[(long)r * ldc] = v;
    }
  }
}

// -------- agent-scope sense-reversing grid barrier --------------------------
__device__ __forceinline__ void grid_barrier(unsigned* cnt, unsigned* gen, unsigned nwg)
{
  __syncthreads();
  if (threadIdx.x == 0) {
    __threadfence();
    unsigned g = __hip_atomic_load(gen, __ATOMIC_RELAXED, __HIP_MEMORY_SCOPE_AGENT);
    unsigned a = __hip_atomic_fetch_add(cnt, 1u, __ATOMIC_ACQ_REL, __HIP_MEMORY_SCOPE_AGENT);
    if (a == nwg - 1u) {
      __hip_atomic_store(cnt, 0u, __ATOMIC_RELAXED, __HIP_MEMORY_SCOPE_AGENT);
      __hip_atomic_fetch_add(gen, 1u, __ATOMIC_RELEASE, __HIP_MEMORY_SCOPE_AGENT);
    } else {
      while (__hip_atomic_load(gen, __ATOMIC_ACQUIRE, __HIP_MEMORY_SCOPE_AGENT) == g) {
        __builtin_amdgcn_s_sleep(2);
      }
    }
  }
  __syncthreads();
}

// -------- persistent LSTM recurrence (64 WGs, 4 waves = 4 gates each) -------
// A tiles (h carries) staged once per WG via async global->LDS, read as ds_b64.
// Faithful to reference bug: carry <- c2 (new cell), cells stay at initial.
__global__ __launch_bounds__(128) void lstm_recurrence(
    const float* __restrict__ hiddens,   // [2,32,512]
    const float* __restrict__ cells,     // [2,32,512]
    const float* __restrict__ W_ih,      // [2,2048,512]
    const float* __restrict__ W_hh,      // [2,2048,512]
    const float* __restrict__ b_ih,      // [2,2048]
    const float* __restrict__ b_hh,      // [2,2048]
    const float* __restrict__ Gx0,       // [T*32, 2048]  (x@W_ih0^T + b0 folded)
    float* __restrict__ h0buf,           // [2][32*512] ping-pong c2 carry L0
    float* __restrict__ h1buf,           // [2][32*512] ping-pong c2 carry L1
    float* __restrict__ h2mid,           // [32*512]  h2 of layer 0 (this step)
    float* __restrict__ s_all,           // [T*32,512] h2 of layer 1 per step
    unsigned* __restrict__ bar, int T_)
{
  const int wg   = blockIdx.x;     // 64 workgroups
  const int mt   = wg >> 5;        // batch tile 0..1
  const int dt   = wg & 31;        // d tile 0..31
  const int wv   = threadIdx.x >> 5;   // gate: 0=i 1=f 2=g 3=o
  const int lane = threadIdx.x & 31;
  const int half = lane >> 4;
  const int lm   = lane & 15;
  const int m0   = mt << 4;
  const int n    = (wv << 9) + (dt << 4);   // gate column base in [0,2048)
  __shared__ __align__(16) float sA[2 * 16 * 516];   // two padded 16x512 A tiles
  __shared__ float ldsg[4 * 256];                    // gate tiles i,f,g,o

  const float* Whh0 = W_hh;
  const float* Whh1 = W_hh + (long)2048 * 512;
  const float* Wih1 = W_ih + (long)2048 * 512;
  const float  bias1 = b_ih[2048 + n + lm] + b_hh[2048 + n + lm];

  const float* bp0  = Whh0 + (long)(n + lm) * 512 + 2 * half;
  const float* bp1x = Wih1 + (long)(n + lm) * 512 + 2 * half;
  const float* bp1h = Whh1 + (long)(n + lm) * 512 + 2 * half;
  const float* aF0  = sA + lm * 516 + 2 * half;            // A-frag base, tile 0
  const float* aF1  = sA + 16 * 516 + lm * 516 + 2 * half; // A-frag base, tile 1
  unsigned* cnt = bar;
  unsigned* gen = bar + 1;

  for (int t = 0; t < T_; ++t) {
    const int w = t & 1;
    const float* h0r = (t == 0) ? hiddens : (h0buf + (long)((t - 1) & 1) * (32 * 512));
    const float* h1r = (t == 0) ? (hiddens + 32 * 512)
                                : (h1buf + (long)((t - 1) & 1) * (32 * 512));

    // ---- layer 0 gates tile: Gx0 tile + h0 @ Whh0^T ----
    stage16x512(h0r + (long)m0 * 512, sA);
    wait_async();
    __syncthreads();
    {
      const float* gx = Gx0 + ((long)t * 32 + m0) * 2048 + n;
      v8f acc[4];
#pragma unroll
      for (int j = 0; j < 4; ++j)
#pragma unroll
        for (int r = 0; r < 8; ++r) acc[j][r] = (j == 0) ? gx[(long)(r + 8 * half) * 2048 + lm] : 0.0f;
      for (int k = 0; k < 512; k += 16) {
#pragma unroll
        for (int j = 0; j < 4; ++j)
          acc[j] = wmma_f32(*(const v2f*)(aF0 + k + 4 * j), *(const v2f*)(bp0 + k + 4 * j), acc[j]);
      }
#pragma unroll
      for (int r = 0; r < 8; ++r)
        ldsg[wv * 256 + (r + 8 * half) * 16 + lm] = acc[0][r] + acc[1][r] + acc[2][r] + acc[3][r];
    }
    __syncthreads();
    for (int e = threadIdx.x; e < 256; e += 128) {
      const int mr = e >> 4, dc = e & 15;
      const int bb = m0 + mr;
      const int d  = (dt << 4) + dc;
      const float ig = ldsg[e], fg = ldsg[256 + e], gg = ldsg[512 + e], og = ldsg[768 + e];
      const float c  = cells[bb * 512 + d];
      const float c2 = sigmf(fg) * c + sigmf(ig) * tanhf(gg);
      const float h2 = sigmf(og) * tanhf(c2);
      h0buf[(long)w * (32 * 512) + bb * 512 + d] = c2;   // carry <- cell (ref bug)
      h2mid[bb * 512 + d] = h2;
    }
    grid_barrier(cnt, gen, 64);

    // ---- layer 1 gates tile: h2_0 @ Wih1^T + h1 @ Whh1^T + biases ----
    stage16x512(h2mid + (long)m0 * 512, sA);
    stage16x512(h1r + (long)m0 * 512, sA + 16 * 516);
    wait_async();
    __syncthreads();
    {
      v8f acc[4];
#pragma unroll
      for (int j = 0; j < 4; ++j)
#pragma unroll
        for (int r = 0; r < 8; ++r) acc[j][r] = (j == 0) ? bias1 : 0.0f;
      for (int k = 0; k < 512; k += 16) {
#pragma unroll
        for (int j = 0; j < 4; ++j)
          acc[j] = wmma_f32(*(const v2f*)(aF0 + k + 4 * j), *(const v2f*)(bp1x + k + 4 * j), acc[j]);
      }
      for (int k = 0; k < 512; k += 16) {
#pragma unroll
        for (int j = 0; j < 4; ++j)
          acc[j] = wmma_f32(*(const v2f*)(aF1 + k + 4 * j), *(const v2f*)(bp1h + k + 4 * j), acc[j]);
      }
#pragma unroll
      for (int r = 0; r < 8; ++r)
        ldsg[wv * 256 + (r + 8 * half) * 16 + lm] = acc[0][r] + acc[1][r] + acc[2][r] + acc[3][r];
    }
    __syncthreads();
    for (int e = threadIdx.x; e < 256; e += 128) {
      const int mr = e >> 4, dc = e & 15;
      const int bb = m0 + mr;
      const int d  = (dt << 4) + dc;
      const float ig = ldsg[e], fg = ldsg[256 + e], gg = ldsg[512 + e], og = ldsg[768 + e];
      const float c  = cells[32 * 512 + bb * 512 + d];
      const float c2 = sigmf(fg) * c + sigmf(ig) * tanhf(gg);
      const float h2 = sigmf(og) * tanhf(c2);
      h1buf[(long)w * (32 * 512) + bb * 512 + d] = c2;   // carry <- cell (ref bug)
      s_all[((long)t * 32 + bb) * 512 + d] = h2;
    }
    grid_barrier(cnt, gen, 64);
  }
}

// -------- masked softmax over S=128, one wave per (t,b) row -----------------
__global__ __launch_bounds__(256) void attn_softmax(
    float* __restrict__ E, const unsigned char* __restrict__ mask, int rows)
{
  const int wv = threadIdx.x >> 5, lane = threadIdx.x & 31;
  const int r = blockIdx.x * (blockDim.x >> 5) + wv;
  if (r >= rows) return;
  const int b = r & 31;                  // r = t*32 + b
  float* e = E + (long)r * 128;
  const unsigned char* mk = mask + b * 128;
  float v[4];
  float mx = -3.0e38f;
#pragma unroll
  for (int i = 0; i < 4; ++i) {
    const int s = i * 32 + lane;
    const float x = mk[s] ? -1.0e9f : e[s];
    v[i] = x;
    mx = fmaxf(mx, x);
  }
#pragma unroll
  for (int off = 16; off >= 1; off >>= 1) mx = fmaxf(mx, __shfl_xor(mx, off, 32));
  float sum = 0.0f;
#pragma unroll
  for (int i = 0; i < 4; ++i) { v[i] = __expf(v[i] - mx); sum += v[i]; }
#pragma unroll
  for (int off = 16; off >= 1; off >>= 1) sum += __shfl_xor(sum, off, 32);
  const float inv = 1.0f / sum;
#pragma unroll
  for (int i = 0; i < 4; ++i) e[i * 32 + lane] = v[i] * inv;
}

// -------- copy s into CAT[:, 1024:1536] -------------------------------------
__global__ __launch_bounds__(128) void copy_s_to_cat(
    const float* __restrict__ s_all, float* __restrict__ CAT)
{
  const int row = blockIdx.x;
  const float4* src = (const float4*)(s_all + (long)row * 512);
  float4* dst = (float4*)(CAT + (long)row * 1536 + 1024);
  for (int i = threadIdx.x; i < 128; i += blockDim.x) dst[i] = src[i];
}

extern "C" void kernel_launch(void* const* d_in, const int* in_sizes, int n_in,
                              void* d_out, int out_size, void* d_ws, size_t ws_size,
                              hipStream_t stream)
{
  const int B = 32, T = 64, S = 128, D = 512, DSRC = 1024;
  const int*   tok     = (const int*)d_in[0];
  const float* enc     = (const float*)d_in[1];
  const float* hiddens = (const float*)d_in[2];
  const float* cells   = (const float*)d_in[3];
  const unsigned char* mask = (const unsigned char*)d_in[4];
  const float* embed   = (const float*)d_in[5];
  const float* W_ih    = (const float*)d_in[6];
  const float* W_hh    = (const float*)d_in[7];
  const float* b_ih    = (const float*)d_in[8];
  const float* b_hh    = (const float*)d_in[9];
  const float* W_in    = (const float*)d_in[10];
  const float* b_in    = (const float*)d_in[11];
  const float* W_out   = (const float*)d_in[12];
  const float* b_out   = (const float*)d_in[13];
  float* out = (float*)d_out;
  (void)in_sizes; (void)n_in; (void)out_size; (void)ws_size;

  char* w = (char*)d_ws;
  auto carve = [&](size_t bytes) -> char* {
    char* p = w; w += (bytes + 255) & ~(size_t)255; return p;
  };
  float* X     = (float*)carve((size_t)T * B * D * 4);          // [T*B, D]
  float* Gx0   = (float*)carve((size_t)T * B * 4 * D * 4);      // [T*B, 4D]
  float* s_all = (float*)carve((size_t)T * B * D * 4);          // [T*B, D]
  float* h0buf = (float*)carve((size_t)2 * B * D * 4);
  float* h1buf = (float*)carve((size_t)2 * B * D * 4);
  float* h2mid = (float*)carve((size_t)B * D * 4);
  float* XQ    = (float*)carve((size_t)T * B * DSRC * 4);       // [T*B, DSRC]
  float* E     = (float*)carve((size_t)T * B * S * 4);          // [T,B,S]
  float* CAT   = (float*)carve((size_t)T * B * (DSRC + D) * 4); // [T*B, 1536]
  unsigned* bar = (unsigned*)carve(256);

  zero_u32<<<1, 32, 0, stream>>>(bar, 8);

  // Phase 0: embedding gather + precompute layer-0 x-contribution for all t
  gather_embed<<<T * B, 128, 0, stream>>>(tok, embed, X, B, T, D);
  {
    const int tiles = (T * B / 64) * (4 * D / 16);               // 4096
    gemm_wmma<<<dim3(tiles / 8, 1, 1), 256, 0, stream>>>(
        X, D, 0, W_ih, D, 0, 0, b_ih, b_hh, Gx0, 4 * D, 0, T * B, 4 * D, D, 0);
  }

  // Phase A: persistent sequential recurrence (64 WGs, grid barriers)
  lstm_recurrence<<<64, 128, 0, stream>>>(hiddens, cells, W_ih, W_hh, b_ih, b_hh,
                                          Gx0, h0buf, h1buf, h2mid, s_all, bar, T);

  // Phase B: attention + output projection over all T*B rows at once
  {
    const int tiles = (T * B / 64) * (DSRC / 16);                // 2048
    gemm_wmma<<<dim3(tiles / 8, 1, 1), 256, 0, stream>>>(
        s_all, D, 0, W_in, D, 0, 0, b_in, nullptr, XQ, DSRC, 0, T * B, DSRC, D, 0);
  }
  {
    const int tiles = (T / 64) * (S / 16);                       // 8 per batch
    gemm_wmma<<<dim3(tiles / 8, 1, B), 256, 0, stream>>>(
        XQ, (long)B * DSRC, DSRC, enc, DSRC, (long)S * DSRC, 0,
        nullptr, nullptr, E, (long)B * S, S, T, S, DSRC, 0);
  }
  attn_softmax<<<(T * B) / 8, 256, 0, stream>>>(E, mask, T * B);
  {
    const int tiles = (T / 64) * (DSRC / 16);                    // 64 per batch
    gemm_wmma<<<dim3(tiles / 8, 1, B), 256, 0, stream>>>(
        E, (long)B * S, S, enc, DSRC, (long)S * DSRC, 1,
        nullptr, nullptr, CAT, (long)B * (DSRC + D), (DSRC + D), T, DSRC, S, 0);
  }
  copy_s_to_cat<<<T * B, 128, 0, stream>>>(s_all, CAT);
  {
    const int tiles = (T * B / 64) * (D / 16);                   // 1024
    gemm_wmma<<<dim3(tiles / 8, 1, 1), 256, 0, stream>>>(
        CAT, DSRC + D, 0, W_out, DSRC + D, 0, 0, b_out, nullptr,
        out, D, 0, T * B, D, DSRC + D, 1);
  }
}